// PosLoss_27393301414259
// MI455X (gfx1250) — compile-verified
//
#include <hip/hip_runtime.h>
#include <math.h>

typedef float v2f __attribute__((ext_vector_type(2)));
typedef float v8f __attribute__((ext_vector_type(8)));

#define NQ     2048
#define VV     6
#define KDIM   4096
#define DD     256
#define ALPHA  0.3f
#define EPSV   1e-6f
#define TOPK   30
#define QSTR   260      // padded LDS row stride for Q tile (bank spread)
#define DSTR   132      // padded LDS row stride for d2 staging
#define CHUNK  128      // keys per chunk (8 waves x 16-key WMMA tiles)
#define NCHUNK (KDIM / CHUNK)   // 32
#define MTILES (NQ / 16)        // 128

__launch_bounds__(256)
__global__ void posloss_main(const float* __restrict__ attn,
                             const float* __restrict__ query,
                             const float* __restrict__ key,
                             const int*   __restrict__ pos_idx,
                             const int*   __restrict__ neg_idx,
                             float*       __restrict__ partials)
{
    __shared__ float q_lds[16 * QSTR];      // 16 queries x 256 dims (padded)
    __shared__ float qq_s[16];              // query squared norms
    __shared__ float d2_s[16 * DSTR];       // 16 x 128 chunk of squared distances
    __shared__ float tv[16][TOPK];          // per-row top-30 values (sorted asc)
    __shared__ int   ti[16][TOPK];          // per-row top-30 key indices
    __shared__ float contrib[16];

    const int tid  = threadIdx.x;
    const int wave = tid >> 5;
    const int lane = tid & 31;
    const int nl   = lane & 15;             // column / row-within-tile
    const int half = lane >> 4;             // 0: K={0,1}, 1: K={2,3} per WMMA step

    const int v  = blockIdx.x / MTILES;
    const int n0 = (blockIdx.x % MTILES) * 16;

    // ---- stage Q tile into LDS (coalesced) ----
    for (int it = 0; it < 16; ++it) {
        int flat = tid + it * 256;          // 0..4095
        int row  = flat >> 8;
        int d    = flat & 255;
        q_lds[row * QSTR + d] =
            query[(size_t)(n0 + row) * (VV * DD) + (size_t)v * DD + d];
    }
    if (tid < 16) {
        for (int j = 0; j < TOPK; ++j) { tv[tid][j] = 3.4e38f; ti[tid][j] = 0; }
    }
    __syncthreads();

    if (tid < 16) {
        float s = 0.f;
        for (int d = 0; d < DD; ++d) { float x = q_lds[tid * QSTR + d]; s += x * x; }
        qq_s[tid] = s;
    }
    __syncthreads();

    // qq values for this lane's 8 accumulator rows (C layout: M = j + 8*half)
    float qqr[8];
    #pragma unroll
    for (int j = 0; j < 8; ++j) qqr[j] = qq_s[j + half * 8];

    const float* keyv = key + (size_t)v * DD;   // key[n*V*D + v*D + d]

    for (int ch = 0; ch < NCHUNK; ++ch) {
        const int kb = ch * CHUNK + wave * 16;  // this wave's key-tile base
        const int kn = kb + nl;                 // this lane's key column
        const float* kptr = keyv + (size_t)kn * (VV * DD) + 2 * half;
        const float* qptr = q_lds + nl * QSTR + 2 * half;

        v8f   acc = {};
        float ks  = 0.f;                        // partial key squared norm
        #pragma unroll 8
        for (int i = 0; i < DD / 4; ++i) {      // 64 chained fp32 WMMAs
            v2f a = *(const v2f*)(qptr + 4 * i);
            v2f b = *(const v2f*)(kptr + 4 * i);
            ks += b.x * b.x + b.y * b.y;
            acc = __builtin_amdgcn_wmma_f32_16x16x4_f32(
                      false, a, false, b, (short)0, acc, false, false);
        }
        // lanes r and r+16 cover complementary dims of the same key
        float kkn = ks + __shfl_xor(ks, 16, 32);

        // d2 = qq + kk - 2*gram (monotone proxy for distance), stage to LDS
        #pragma unroll
        for (int j = 0; j < 8; ++j) {
            float d2 = fmaxf(qqr[j] + kkn - 2.0f * acc[j], 0.0f);
            d2_s[(j + half * 8) * DSTR + wave * 16 + nl] = d2;
        }
        __syncthreads();

        // stable insertion into per-row top-30 (ascending value, tie -> lower idx)
        if (tid < 16) {
            const int m = tid;
            float worst = tv[m][TOPK - 1];
            for (int c = 0; c < CHUNK; ++c) {
                float d2 = d2_s[m * DSTR + c];
                if (d2 < worst) {
                    int kidx = ch * CHUNK + c;
                    int p = TOPK - 1;
                    while (p > 0 && tv[m][p - 1] > d2) {
                        tv[m][p] = tv[m][p - 1];
                        ti[m][p] = ti[m][p - 1];
                        --p;
                    }
                    tv[m][p] = d2;
                    ti[m][p] = kidx;
                    worst = tv[m][TOPK - 1];
                }
            }
        }
        __syncthreads();
    }

    // ---- epilogue: triplet-margin terms for this WG's 16 rows ----
    if (tid < 16) {
        const int m  = tid;
        const int nq = n0 + m;
        const int r  = v * NQ + nq;             // row id in reference's (V*Nq) flatten
        const int pi = pos_idx[r];
        const int ni = neg_idx[r];
        const int kp = ti[m][pi];
        const int kn2 = ti[m][ni];
        const size_t base = (size_t)nq * (VV * KDIM) + (size_t)v * KDIM;
        const float ap = attn[base + kp];
        const float an = attn[base + kn2];
        const float pos = 1.0f / (1.0f + __expf(-ap));
        const float neg = 1.0f / (1.0f + __expf(-an));
        const float dp = fabsf(1.0f - pos + EPSV);
        const float dn = fabsf(1.0f - neg + EPSV);
        contrib[m] = fmaxf(dp - dn + ALPHA, 0.0f);
    }
    __syncthreads();
    if (tid == 0) {
        float s = 0.f;
        for (int m = 0; m < 16; ++m) s += contrib[m];
        partials[blockIdx.x] = s;
    }
}

__global__ void posloss_reduce(const float* __restrict__ partials,
                               float* __restrict__ out, int n)
{
    if (blockIdx.x == 0 && threadIdx.x == 0) {
        float s = 0.f;
        for (int i = 0; i < n; ++i) s += partials[i];   // fixed order: deterministic
        out[0] = s * (1.0f / (float)(VV * NQ));          // LOSS_WEIGHT = 1
    }
}

extern "C" void kernel_launch(void* const* d_in, const int* in_sizes, int n_in,
                              void* d_out, int out_size, void* d_ws, size_t ws_size,
                              hipStream_t stream)
{
    (void)in_sizes; (void)n_in; (void)out_size; (void)ws_size;
    const float* attn    = (const float*)d_in[0];
    // d_in[1] = query2gt: validity check is always-true by construction -> unused
    const float* query   = (const float*)d_in[2];
    const float* key     = (const float*)d_in[3];
    const int*   pos_idx = (const int*)d_in[4];
    const int*   neg_idx = (const int*)d_in[5];

    float* partials = (float*)d_ws;          // 768 floats
    float* out      = (float*)d_out;

    const int nblocks = VV * MTILES;         // 768
    posloss_main<<<nblocks, 256, 0, stream>>>(attn, query, key, pos_idx, neg_idx, partials);
    posloss_reduce<<<1, 32, 0, stream>>>(partials, out, nblocks);
}